// Resolutionaware_Multiscale_Progressive_Attention_70334384439844
// MI455X (gfx1250) — compile-verified
//
#include <hip/hip_runtime.h>
#include <hip/hip_bf16.h>
#include <cstdint>

// Problem constants (match reference setup_inputs)
#define BB   4
#define NPTS 65536
#define CC   256
#define HH   200
#define WW   176
#define HWW  (HH * WW)          // 35200 = 32 * 1100

typedef __attribute__((ext_vector_type(2))) float v2f;
typedef __attribute__((ext_vector_type(8))) float v8f;

// ---------------------------------------------------------------------------
// Kernel 1: transpose bev (B, C, H*W) -> bevT (B, H*W, C). 32x32 LDS tiles.
// bev is read exactly once -> non-temporal loads keep L2 free for bevT.
// ---------------------------------------------------------------------------
__global__ void bev_transpose_kernel(const float* __restrict__ in,
                                     float* __restrict__ outT) {
    __shared__ float tile[32][33];
    const int b  = blockIdx.z;
    const int c0 = blockIdx.y * 32;       // C/32 = 8 tiles, exact
    const int p0 = blockIdx.x * 32;       // HW/32 = 1100 tiles, exact
    const int tx = threadIdx.x;           // 0..31
    const int ty = threadIdx.y;           // 0..7

#pragma unroll
    for (int i = 0; i < 32; i += 8) {
        const int c = c0 + ty + i;
        const int p = p0 + tx;            // consecutive -> coalesced read
        tile[ty + i][tx] =
            __builtin_nontemporal_load(&in[((size_t)b * CC + c) * HWW + p]);
    }
    __syncthreads();
#pragma unroll
    for (int i = 0; i < 32; i += 8) {
        const int p = p0 + ty + i;
        const int c = c0 + tx;            // consecutive -> coalesced write
        outT[((size_t)b * HWW + p) * CC + c] = tile[tx][ty + i];
    }
}

// ---------------------------------------------------------------------------
// Kernel 2: WMMA bilinear interpolation from bevT (B, H, W, C).
// One wave32 handles 16 keypoints. Out(16kp x 16ch) accumulated via 16 chained
// V_WMMA_F32_16X16X4_F32 ops (block-diagonal weight matrix, full f32 precision).
//
// A (16x4 f32, 2 VGPR): lanes 0-15 -> M=lane, K={0,1}; lanes 16-31 -> K={2,3}
// B (4x16 f32, 2 VGPR): lanes 0-15 -> N=lane, K={0,1}; lanes 16-31 -> K={2,3}
// C/D (16x16 f32, 8 VGPR): VGPR r, lane l -> M = r + 8*(l>=16), N = l&15
// ---------------------------------------------------------------------------
__global__ void
__launch_bounds__(256, 2)
bev_interp_wmma_kernel(const float* __restrict__ keypoints,
                       const float* __restrict__ bevT,
                       float* __restrict__ out) {
    __shared__ int   s_off[8][16][4];   // per-wave corner element offsets
    __shared__ float s_w[8][16][4];     // per-wave corner weights

    const int lane = threadIdx.x & 31;
    const int wv   = threadIdx.x >> 5;                 // 0..7
    const int tileId = blockIdx.x * 8 + wv;            // 0..16383
    const int b   = tileId / (NPTS / 16);
    const int n0  = (tileId % (NPTS / 16)) * 16;

    // ---- setup: lanes 0..15 each own one keypoint -------------------------
    if (lane < 16) {
        const int j = lane;
        const size_t kbase = ((size_t)b * NPTS + n0 + j) * 3;
        const float kx = keypoints[kbase + 0];
        const float ky = keypoints[kbase + 1];
        // match reference: (k - range) / voxel / stride
        const float x = (kx - 0.0f)   / 0.05f / 8.0f;
        const float y = (ky + 40.0f)  / 0.05f / 8.0f;
        const int x0 = (int)floorf(x);
        const int y0 = (int)floorf(y);
        const int x0c = min(max(x0, 0), WW - 1);
        const int x1c = min(max(x0 + 1, 0), WW - 1);
        const int y0c = min(max(y0, 0), HH - 1);
        const int y1c = min(max(y0 + 1, 0), HH - 1);
        const float fx0 = (float)x0c, fx1 = (float)x1c;
        const float fy0 = (float)y0c, fy1 = (float)y1c;
        // corner order: a=(y0,x0) b=(y1,x0) c=(y0,x1) d=(y1,x1)
        s_w[wv][j][0] = (fx1 - x) * (fy1 - y);
        s_w[wv][j][1] = (fx1 - x) * (y - fy0);
        s_w[wv][j][2] = (x - fx0) * (fy1 - y);
        s_w[wv][j][3] = (x - fx0) * (y - fy0);
        s_off[wv][j][0] = (y0c * WW + x0c) * CC;
        s_off[wv][j][1] = (y1c * WW + x0c) * CC;
        s_off[wv][j][2] = (y0c * WW + x1c) * CC;
        s_off[wv][j][3] = (y1c * WW + x1c) * CC;
    }
    __syncthreads();

    const int nl    = lane & 15;          // N / channel sub-index
    const int khalf = (lane >> 4) << 1;   // 0 for low half (K0,K1), 2 for high (K2,K3)
    const bool selRow = true;

    // A-matrix values are channel-tile invariant: precompute into registers.
    float a0v[16], a1v[16];
    int   o0v[16], o1v[16];
#pragma unroll
    for (int j = 0; j < 16; ++j) {
        const float w0 = s_w[wv][j][khalf];
        const float w1 = s_w[wv][j][khalf + 1];
        const bool sel = (nl == j);
        a0v[j] = sel ? w0 : 0.0f;
        a1v[j] = sel ? w1 : 0.0f;
        o0v[j] = s_off[wv][j][khalf];
        o1v[j] = s_off[wv][j][khalf + 1];
    }

    const float* __restrict__ bevT_b = bevT + (size_t)b * HWW * CC;
    const size_t outRowBase = ((size_t)b * NPTS + n0) * CC;
    const int mOfs = (lane >> 4) * 8;

    for (int t = 0; t < 16; ++t) {        // 16 channel tiles of 16
        const int c0 = t * 16;
        v8f acc = {};
#pragma unroll
        for (int j = 0; j < 16; ++j) {
            const float b0 = bevT_b[o0v[j] + c0 + nl];   // 64B contiguous / half-wave
            const float b1 = bevT_b[o1v[j] + c0 + nl];
            v2f A = {a0v[j], a1v[j]};
            v2f Bv = {b0, b1};
            acc = __builtin_amdgcn_wmma_f32_16x16x4_f32(
                false, A, false, Bv, (short)0, acc, false, false);
        }
        // scatter D per 16x16 f32 C/D layout; output is write-once -> NT stores
#pragma unroll
        for (int r = 0; r < 8; ++r) {
            const int m = mOfs + r;
            __builtin_nontemporal_store(
                acc[r], &out[outRowBase + (size_t)m * CC + c0 + nl]);
        }
    }
    (void)selRow;
}

// ---------------------------------------------------------------------------
// Fallback (workspace too small for transposed tensor): direct strided gather
// from (B, C, H, W). One block per keypoint, one thread per channel.
// ---------------------------------------------------------------------------
__global__ void bev_interp_direct_kernel(const float* __restrict__ keypoints,
                                         const float* __restrict__ bev,
                                         float* __restrict__ out) {
    const int flat = blockIdx.x;                  // 0 .. B*NPTS-1
    const int b = flat / NPTS;
    const int n = flat % NPTS;
    const int c = threadIdx.x;                    // 0..255

    const size_t kbase = ((size_t)b * NPTS + n) * 3;
    const float kx = keypoints[kbase + 0];
    const float ky = keypoints[kbase + 1];
    const float x = (kx - 0.0f)  / 0.05f / 8.0f;
    const float y = (ky + 40.0f) / 0.05f / 8.0f;
    const int x0 = (int)floorf(x);
    const int y0 = (int)floorf(y);
    const int x0c = min(max(x0, 0), WW - 1);
    const int x1c = min(max(x0 + 1, 0), WW - 1);
    const int y0c = min(max(y0, 0), HH - 1);
    const int y1c = min(max(y0 + 1, 0), HH - 1);
    const float fx0 = (float)x0c, fx1 = (float)x1c;
    const float fy0 = (float)y0c, fy1 = (float)y1c;
    const float wa = (fx1 - x) * (fy1 - y);
    const float wb = (fx1 - x) * (y - fy0);
    const float wc = (x - fx0) * (fy1 - y);
    const float wd = (x - fx0) * (y - fy0);

    const float* plane = bev + ((size_t)b * CC + c) * HWW;
    const float Ia = plane[y0c * WW + x0c];
    const float Ib = plane[y1c * WW + x0c];
    const float Ic = plane[y0c * WW + x1c];
    const float Id = plane[y1c * WW + x1c];
    __builtin_nontemporal_store(Ia * wa + Ib * wb + Ic * wc + Id * wd,
                                &out[((size_t)b * NPTS + n) * CC + c]);
}

// ---------------------------------------------------------------------------
extern "C" void kernel_launch(void* const* d_in, const int* in_sizes, int n_in,
                              void* d_out, int out_size, void* d_ws, size_t ws_size,
                              hipStream_t stream) {
    const float* keypoints = (const float*)d_in[0];   // (B, N, 3) f32
    const float* bev       = (const float*)d_in[1];   // (B, C, H, W) f32
    float* out             = (float*)d_out;           // (B, N, C) f32

    const size_t needT = (size_t)BB * HWW * CC * sizeof(float);   // 144 MB

    if (ws_size >= needT) {
        float* bevT = (float*)d_ws;
        dim3 tgrid(HWW / 32, CC / 32, BB);            // (1100, 8, 4)
        dim3 tblk(32, 8);
        bev_transpose_kernel<<<tgrid, tblk, 0, stream>>>(bev, bevT);

        const int tiles  = BB * (NPTS / 16);          // 16384
        const int blocks = tiles / 8;                 // 2048 (8 waves/block)
        bev_interp_wmma_kernel<<<blocks, 256, 0, stream>>>(keypoints, bevT, out);
    } else {
        bev_interp_direct_kernel<<<BB * NPTS, CC, 0, stream>>>(keypoints, bev, out);
    }
}